// CodebookLayer_54073638256732
// MI455X (gfx1250) — compile-verified
//
#include <hip/hip_runtime.h>

// VQ snap kernel for MI455X (gfx1250, wave32, WMMA).
//
// logits = X[16384,1024] * CB^T[1024,8192]; argmax over codes; gather rows.
// Memory-bound (~168 MB min HBM traffic), and L2-traffic-bound above that:
// each WG streams the L2-resident 33.5 MB codebook once, so total L2 reads
// = (16384/M_TILE) * 33.5 MB. M_TILE=64 maxes x-reuse at the 320 KB/WGP LDS
// capacity (256 KB x tile), minimizing that stream at ~8.6 GB. f32 WMMA
// (V_WMMA_F32_16X16X4_F32) reproduces the reference arithmetic exactly and
// its compute time (<1us) is far below the memory floor, so there is no
// reason to downcast (bf16 inputs could flip near-tie argmax winners).
//
// One WG/WGP (2 waves/SIMD) can't hide L2 latency via occupancy, so the
// K-loop is unrolled 8x: ~16 codebook b64 loads + 4 fused LDS A-loads in
// flight per wave per iteration (long s_clause), overlapping the WMMAs.

typedef float v2f __attribute__((ext_vector_type(2)));
typedef float v8f __attribute__((ext_vector_type(8)));

#define DIM        1024
#define NUM_CODES  8192
#define ROWS       (8 * 2048)

#define M_TILE     64                 // x rows per workgroup
#define XS_STRIDE  (DIM + 4)          // +16B row pad: lanes land on distinct LDS banks
#define NW_N       2                  // waves along N
#define N_CHUNK    64                 // codes per outer iteration (2 waves * 2 subtiles * 16)

// dynamic LDS: x tile | reduce vals | reduce idxs | final idx
#define SMEM_FLOATS (M_TILE * XS_STRIDE + M_TILE * NW_N)
#define SMEM_INTS   (M_TILE * NW_N + M_TILE)
#define SMEM_BYTES  ((SMEM_FLOATS + SMEM_INTS) * 4)

__global__ __launch_bounds__(256)
void vq_snap_kernel(const float* __restrict__ x,
                    const float* __restrict__ cb,
                    float* __restrict__ out) {
    extern __shared__ char smem_raw[];
    float* xs    = (float*)smem_raw;                       // [M_TILE][XS_STRIDE]
    float* red_v = xs + M_TILE * XS_STRIDE;                // [M_TILE][NW_N]
    int*   red_i = (int*)(red_v + M_TILE * NW_N);          // [M_TILE][NW_N]
    int*   fin_i = red_i + M_TILE * NW_N;                  // [M_TILE]

    const int tid = threadIdx.x;
    const int m0  = blockIdx.x * M_TILE;

    // ---- stage x tile into LDS (coalesced float4) ----
    {
        const float4* xg = (const float4*)(x + (size_t)m0 * DIM);
        for (int i = tid; i < M_TILE * (DIM / 4); i += 256) {
            const int r = i / (DIM / 4), c = i % (DIM / 4);
            const float4 v = xg[(size_t)r * (DIM / 4) + c];
            float* dst = xs + r * XS_STRIDE + c * 4;
            dst[0] = v.x; dst[1] = v.y; dst[2] = v.z; dst[3] = v.w;
        }
    }
    __syncthreads();

    const int wave  = tid >> 5;
    const int lane  = tid & 31;
    const int hh    = lane >> 4;       // wave half (K pairing per ISA 16x4 f32 layout)
    const int l16   = lane & 15;
    const int m_sub = wave & 3;        // 0..3 : which 16-row A subtile
    const int n_sub = wave >> 2;       // 0..1 : which 32-code B slice

    // A operand per lane: v2f at x[m_sub*16 + l16][k + 2*hh]
    const float* xrow = xs + (m_sub * 16 + l16) * XS_STRIDE + 2 * hh;

    float best_v0[8], best_v1[8];
    int   best_i0[8], best_i1[8];
#pragma unroll
    for (int r = 0; r < 8; ++r) {
        best_v0[r] = -__builtin_inff(); best_i0[r] = 0;
        best_v1[r] = -__builtin_inff(); best_i1[r] = 0;
    }

    // lane's two code columns within a chunk
    const int col0_off = n_sub * 32 + l16;       // subtile 0
    const int col1_off = col0_off + 16;          // subtile 1

    for (int n0 = 0; n0 < NUM_CODES; n0 += N_CHUNK) {
        const int c0 = n0 + col0_off;
        const int c1 = n0 + col1_off;
        // B operand per lane: v2f at cb[c][k + 2*hh] (B rows striped across lanes)
        const float* brow0 = cb + (size_t)c0 * DIM + 2 * hh;
        const float* brow1 = cb + (size_t)c1 * DIM + 2 * hh;

        // prefetch 2 chunks ahead (L2-resident; emits global_prefetch_b8)
        if (n0 + 2 * N_CHUNK < NUM_CODES) {
            __builtin_prefetch(cb + (size_t)(c0 + 2 * N_CHUNK) * DIM, 0, 1);
            __builtin_prefetch(cb + (size_t)(c1 + 2 * N_CHUNK) * DIM, 0, 1);
        }

        v8f acc0 = {};
        v8f acc1 = {};
#pragma unroll 8
        for (int k = 0; k < DIM; k += 4) {
            const v2f a  = *(const v2f*)(xrow + k);
            const v2f b0 = *(const v2f*)(brow0 + k);
            const v2f b1 = *(const v2f*)(brow1 + k);
            // D = A x B + C, f32 16x16x4 (exact reference arithmetic)
            acc0 = __builtin_amdgcn_wmma_f32_16x16x4_f32(
                false, a, false, b0, (short)0, acc0, false, false);
            acc1 = __builtin_amdgcn_wmma_f32_16x16x4_f32(
                false, a, false, b1, (short)0, acc1, false, false);
        }

        // running argmax per lane; n strictly increases per lane across chunks,
        // so '>' preserves first-occurrence (jnp.argmax) semantics.
#pragma unroll
        for (int r = 0; r < 8; ++r) {
            if (acc0[r] > best_v0[r]) { best_v0[r] = acc0[r]; best_i0[r] = c0; }
            if (acc1[r] > best_v1[r]) { best_v1[r] = acc1[r]; best_i1[r] = c1; }
        }
    }

    // ---- reduce: combine both subtiles per lane, then across the 16-lane half
    // C layout: VGPR r, lane L -> row M = r + 8*(L>>4), col N = L&15.
#pragma unroll
    for (int r = 0; r < 8; ++r) {
        float bv; int bi;
        {
            const float v0 = best_v0[r], v1 = best_v1[r];
            const int   i0 = best_i0[r], i1 = best_i1[r];
            const bool take1 = (v1 > v0) || (v1 == v0 && i1 < i0);
            bv = take1 ? v1 : v0;
            bi = take1 ? i1 : i0;
        }
#pragma unroll
        for (int off = 1; off < 16; off <<= 1) {   // 1,2,4,8 stay within the half
            const float ov = __shfl_xor(bv, off, 32);
            const int   oi = __shfl_xor(bi, off, 32);
            if (ov > bv || (ov == bv && oi < bi)) { bv = ov; bi = oi; }
        }
        if (l16 == 0) {
            const int m_local = m_sub * 16 + r + 8 * hh;
            red_v[m_local * NW_N + n_sub] = bv;
            red_i[m_local * NW_N + n_sub] = bi;
        }
    }
    __syncthreads();

    // ---- combine across the NW_N wave groups
    if (tid < M_TILE) {
        float bv = red_v[tid * NW_N + 0];
        int   bi = red_i[tid * NW_N + 0];
#pragma unroll
        for (int j = 1; j < NW_N; ++j) {
            const float ov = red_v[tid * NW_N + j];
            const int   oi = red_i[tid * NW_N + j];
            if (ov > bv || (ov == bv && oi < bi)) { bv = ov; bi = oi; }
        }
        fin_i[tid] = bi;
    }
    __syncthreads();

    // ---- gather: out[m] = codebook[argmax[m]], coalesced float4
    for (int r = 0; r < M_TILE; ++r) {
        const int code = fin_i[r];
        const float4* src = (const float4*)(cb + (size_t)code * DIM);
        float4* dst = (float4*)(out + (size_t)(m0 + r) * DIM);
        dst[tid] = src[tid];   // DIM/4 == 256 == blockDim.x
    }
}

extern "C" void kernel_launch(void* const* d_in, const int* in_sizes, int n_in,
                              void* d_out, int out_size, void* d_ws, size_t ws_size,
                              hipStream_t stream) {
    (void)in_sizes; (void)n_in; (void)out_size; (void)d_ws; (void)ws_size;
    const float* x  = (const float*)d_in[0];
    const float* cb = (const float*)d_in[1];
    float* out      = (float*)d_out;

    // LDS > default 64KB dynamic limit: opt in (deterministic, capture-safe)
    hipFuncSetAttribute((const void*)vq_snap_kernel,
                        hipFuncAttributeMaxDynamicSharedMemorySize, SMEM_BYTES);

    vq_snap_kernel<<<ROWS / M_TILE, 256, SMEM_BYTES, stream>>>(x, cb, out);
}